// BagModel_3d_6536940225208
// MI455X (gfx1250) — compile-verified
//
#include <hip/hip_runtime.h>

// CDNA5 (gfx1250) fused BagModel kernel:
//   out[b] = (1/n_b) * sum_{n < n_b} relu(x[b,n,:] @ W1 + b1) @ W2 + b2
//
// MI455X strategy:
//  - fp32 WMMA (v_wmma_f32_16x16x4_f32): reference is fp32; at 17.2 GFLOP over
//    128 MB the kernel is fp32-matrix-compute bound -> maximize WMMA issue and
//    keep all operands LDS-resident.
//  - Entire W1 (256KB) in LDS (320KB/WGP on CDNA5), interleaved K-pair layout
//    -> each B fragment is a single ds_load_b64; pair-row pitch 272 puts the
//    two half-wave groups on disjoint bank sets.
//  - x tiles staged via CDNA5 async-tensor path: global_load_async_to_lds_b128
//    (ASYNCcnt) with double buffering; tile t+1 copy overlaps tile t compute.
//  - LDS layout: xt FIRST (low offsets) so all A-fragment addresses fit the
//    16-bit DS immediate field (enables ds_load_2addr_b64 / offset folding);
//    the 278KB W1 region, which needs a rolling base register anyway, sits above.
//  - Epilogue fuses bias + ReLU + dot(W2) + ragged mask; one scalar per bag
//    leaves the WGP.

typedef float v2f __attribute__((ext_vector_type(2)));
typedef float v8f __attribute__((ext_vector_type(8)));

#define NBAGS 256
#define NMAX  512
#define DIN   256
#define DH    256
#define DHP   272   // padded pair-row pitch (in pairs) for W1 in LDS
#define XTP   260   // padded x-tile row pitch (floats); 260 % 64 == 4 -> conflict-free A reads
#define XTSZ  (16 * XTP)
#define W1OFF (2 * XTSZ + 8)   // float offset of w1p; 8328 floats -> 33312 B, 8B-aligned

// Async copy: 16B global -> LDS, same immediate offset applied to both sides.
__device__ __forceinline__ void async_cp_b128(uint32_t lds_dst, const float* gsrc, int ioff) {
    asm volatile("global_load_async_to_lds_b128 %0, %1, off offset:%2"
                 :: "v"(lds_dst), "v"(gsrc), "i"(ioff) : "memory");
}
#define S_WAIT_ASYNCCNT(n) asm volatile("s_wait_asynccnt " #n ::: "memory")

// Generic->LDS byte offset (low 32 bits of the shared-aperture address).
__device__ __forceinline__ uint32_t lds_addr(const void* p) {
    return (uint32_t)(uintptr_t)p;
}

__global__ __launch_bounds__(256) void bag_model_wmma_f32(
    const float* __restrict__ x,      // [NBAGS, NMAX, DIN]
    const int*   __restrict__ nInst,  // [NBAGS]
    const float* __restrict__ W1,     // [DIN, DH]
    const float* __restrict__ b1,     // [DH]
    const float* __restrict__ W2,     // [DH] (x1)
    const float* __restrict__ b2,     // [1]
    float*       __restrict__ out)    // [NBAGS]
{
    extern __shared__ float smem[];
    float* xt   = smem;                 // [2][16][XTP] double-buffered x tile (LOW offsets)
    float* accp = smem + 2 * XTSZ;      // [1] bag accumulator
    float* w1p  = smem + W1OFF;         // [DIN/2][DHP][2] interleaved K-pairs (HIGH offsets)

    const int tid  = threadIdx.x;
    const int bag  = blockIdx.x;
    const int lane = tid & 31;
    const int wv   = tid >> 5;      // wave id 0..7
    const int hi   = lane >> 4;     // half-wave select (K/M split per ISA layout)
    const int l16  = lane & 15;

    const int nb     = nInst[bag];
    const int ntiles = (nb + 15) >> 4;

    const float* xb = x + (size_t)bag * NMAX * DIN;

    // ---- per-thread staging coordinates (16 x 256 tile, 4 x b128 per thread)
    const int sr = tid >> 4;             // tile row 0..15
    const int sc = (tid & 15) * 16;      // float column 0..240

    // ---- kick off async copies of tiles 0 and 1 (direct to LDS, ASYNCcnt)
    {
        const uint32_t d0 = lds_addr(&xt[sr * XTP + sc]);
        const float*   s0 = xb + (size_t)sr * DIN + sc;
        async_cp_b128(d0, s0,  0);
        async_cp_b128(d0, s0, 16);
        async_cp_b128(d0, s0, 32);
        async_cp_b128(d0, s0, 48);
        if (ntiles > 1) {
            const uint32_t d1 = d0 + XTSZ * sizeof(float);
            const float*   s1 = s0 + 16 * DIN;
            async_cp_b128(d1, s1,  0);
            async_cp_b128(d1, s1, 16);
            async_cp_b128(d1, s1, 32);
            async_cp_b128(d1, s1, 48);
        }
    }

    // ---- stage W1 into LDS as K-pairs: w1p[k2][n] = (W1[2k2][n], W1[2k2+1][n])
    for (int p = tid; p < (DIN / 2) * DH; p += 256) {
        const int k2 = p >> 8;      // 0..127
        const int n  = p & 255;
        v2f pr;
        pr.x = W1[(2 * k2) * DH + n];
        pr.y = W1[(2 * k2 + 1) * DH + n];
        *(v2f*)&w1p[(k2 * DHP + n) * 2] = pr;
    }
    if (tid == 0) accp[0] = 0.0f;

    // per-lane epilogue constants for this wave's two 16-column tiles
    const int   col0 = wv * 32 + l16;
    const int   col1 = col0 + 16;
    const float bia0 = b1[col0], bia1 = b1[col1];
    const float w20  = W2[col0], w21  = W2[col1];

    float part = 0.0f;

    for (int t = 0; t < ntiles; ++t) {
        // wait for tile t's async copy (leave tile t+1's 4 ops in flight)
        if (t + 1 < ntiles) { S_WAIT_ASYNCCNT(4); }
        else                { S_WAIT_ASYNCCNT(0); }
        __syncthreads();

        const float* xbuf = xt + (t & 1) * XTSZ;

        // ---- K loop: D_IN = 256 -> 64 steps of K=4, two WMMAs per step
        v8f c0 = {};
        v8f c1 = {};
#pragma unroll 8
        for (int kk = 0; kk < DIN / 4; ++kk) {
            // A fragment (16x4 f32): lanes 0-15 K=4kk..4kk+1, lanes 16-31 K=4kk+2..4kk+3
            const int kb = kk * 4 + 2 * hi;
            v2f a = *(const v2f*)&xbuf[l16 * XTP + kb];
            // B fragments (4x16 f32): pair (K, K+1) per lane, one b64 each
            const int k2 = kk * 2 + hi;
            v2f bf0 = *(const v2f*)&w1p[(k2 * DHP + col0) * 2];
            v2f bf1 = *(const v2f*)&w1p[(k2 * DHP + col1) * 2];
            c0 = __builtin_amdgcn_wmma_f32_16x16x4_f32(
                     false, a, false, bf0, (short)0, c0, false, false);
            c1 = __builtin_amdgcn_wmma_f32_16x16x4_f32(
                     false, a, false, bf1, (short)0, c1, false, false);
        }

        // ---- fused epilogue: bias + ReLU + dot(W2) + ragged validity mask
        // C layout: VGPR r <-> M = r + 8*hi ; N = colbase + l16
        const int rowbase = t * 16 + hi * 8;
#pragma unroll
        for (int r = 0; r < 8; ++r) {
            const float m  = (rowbase + r < nb) ? 1.0f : 0.0f;
            float z0 = c0[r] + bia0; z0 = z0 > 0.0f ? z0 : 0.0f;
            float z1 = c1[r] + bia1; z1 = z1 > 0.0f ? z1 : 0.0f;
            part += m * (z0 * w20 + z1 * w21);
        }
        __syncthreads();   // all waves done reading buf (t&1) before refilling it

        // ---- refill the buffer we just consumed with tile t+2
        if (t + 2 < ntiles) {
            const uint32_t d = lds_addr(&xt[(t & 1) * XTSZ + sr * XTP + sc]);
            const float*   s = xb + (size_t)((t + 2) * 16 + sr) * DIN + sc;
            async_cp_b128(d, s,  0);
            async_cp_b128(d, s, 16);
            async_cp_b128(d, s, 32);
            async_cp_b128(d, s, 48);
        }
    }

    atomicAdd(accp, part);   // ds_add_f32
    __syncthreads();
    if (tid == 0) out[bag] = accp[0] / (float)nb + b2[0];
}

extern "C" void kernel_launch(void* const* d_in, const int* in_sizes, int n_in,
                              void* d_out, int out_size, void* d_ws, size_t ws_size,
                              hipStream_t stream) {
    const float* x     = (const float*)d_in[0];
    const int*   nInst = (const int*)d_in[1];
    const float* W1    = (const float*)d_in[2];
    const float* b1    = (const float*)d_in[3];
    const float* W2    = (const float*)d_in[4];
    const float* b2    = (const float*)d_in[5];
    float*       out   = (float*)d_out;

    const size_t shmem =
        (size_t)(W1OFF + 2 * (DIN / 2) * DHP) * sizeof(float); // ~305 KB < 320 KB/WGP

    // Allow >64KB dynamic LDS (CDNA5 WGP has 320KB). Idempotent, capture-safe.
    hipFuncSetAttribute((const void*)bag_model_wmma_f32,
                        hipFuncAttributeMaxDynamicSharedMemorySize,
                        (int)shmem);

    bag_model_wmma_f32<<<NBAGS, 256, shmem, stream>>>(x, nInst, W1, b1, W2, b2, out);
}